// LightGCN_7129645711540
// MI455X (gfx1250) — compile-verified
//
#include <hip/hip_runtime.h>

#define D 64
#define TILE 512
#define SCATTER_BLOCK 256
#define WAVES_PER_BLOCK (SCATTER_BLOCK / 32)

// ---- CDNA5 async global->LDS path (probe via __has_builtin; safe fallback) ----
#if defined(__has_builtin)
#if __has_builtin(__builtin_amdgcn_global_load_async_to_lds_b64) && \
    __has_builtin(__builtin_amdgcn_s_wait_asynccnt)
#define HAVE_ASYNC_LDS 1
#endif
#endif
#ifndef HAVE_ASYNC_LDS
#define HAVE_ASYNC_LDS 0
#endif

#if HAVE_ASYNC_LDS
typedef int v2i __attribute__((ext_vector_type(2)));
typedef __attribute__((address_space(1))) v2i gv2i;   // global v2i
typedef __attribute__((address_space(3))) v2i lv2i;   // LDS v2i
__device__ __forceinline__ void async_copy_b64(const int* g, int* l) {
  // GLOBAL_LOAD_ASYNC_TO_LDS_B64: 2 dwords, tracked by ASYNCcnt
  __builtin_amdgcn_global_load_async_to_lds_b64((gv2i*)g, (lv2i*)l, 0, 0);
}
__device__ __forceinline__ void wait_async_le2() { __builtin_amdgcn_s_wait_asynccnt(2); }
__device__ __forceinline__ void wait_async_le0() { __builtin_amdgcn_s_wait_asynccnt(0); }
#endif

// -------------------- main scatter-add kernel (first so it leads the disasm) ----
// dst[edst[e]] += (rdeg ? rdeg[esrc[e]] : 1) * src[esrc[e]]   (rows of D floats)
// Edge indices are staged tile-by-tile into LDS via async global->LDS loads,
// double-buffered and synchronized with s_wait_asynccnt. One wave per edge:
// lane l handles cols [2l, 2l+1] (float2 gather, 2 wave-wide f32 atomics).
__global__ __launch_bounds__(SCATTER_BLOCK) void scatter_kernel(
    const float* __restrict__ src, const float* __restrict__ rdeg,
    const int* __restrict__ esrc, const int* __restrict__ edst,
    float* __restrict__ dst, int E) {
  __shared__ int lsrc[2][TILE];
  __shared__ int ldst[2][TILE];
  const int tid  = threadIdx.x;
  const int wave = tid >> 5;
  const int lane = tid & 31;
  const int ntiles = (E + TILE - 1) / TILE;
  const int stride = gridDim.x;

  // stages tile t into buffer b; returns true if async (full tile) staging used
  auto stage = [&](int t, int b) -> bool {
    const int base = t * TILE;
    const int off = tid * 2;  // 256 threads * 2 dwords = 512 = TILE
    if (base + TILE <= E) {
#if HAVE_ASYNC_LDS
      async_copy_b64(esrc + base + off, &lsrc[b][off]);
      async_copy_b64(edst + base + off, &ldst[b][off]);
      return true;
#else
      lsrc[b][off]     = esrc[base + off];
      lsrc[b][off + 1] = esrc[base + off + 1];
      ldst[b][off]     = edst[base + off];
      ldst[b][off + 1] = edst[base + off + 1];
      return false;
#endif
    } else {
      for (int k = 0; k < 2; ++k) {
        int j = off + k;
        if (base + j < E) {
          lsrc[b][j] = esrc[base + j];
          ldst[b][j] = edst[base + j];
        }
      }
      return false;
    }
  };

  int t = (int)blockIdx.x;
  if (t < ntiles) stage(t, 0);
  int buf = 0;
  for (; t < ntiles; t += stride) {
    const int tn = t + stride;
    bool next_async = false;
    if (tn < ntiles) next_async = stage(tn, buf ^ 1);
#if HAVE_ASYNC_LDS
    // Ensure tile t's async loads are complete: if the next tile was also
    // async-staged there are exactly 2 newer outstanding ops per wave;
    // otherwise drain to zero (covers tail-tile staging done with plain stores).
    if (next_async) wait_async_le2();
    else            wait_async_le0();
#else
    (void)next_async;
#endif
    __syncthreads();

    const int base = t * TILE;
    const int jmax = min(TILE, E - base);
    for (int j = wave; j < jmax; j += WAVES_PER_BLOCK) {
      const int us = lsrc[buf][j];
      const int ud = ldst[buf][j];
      float2 v = ((const float2*)(src + (long long)us * D))[lane];
      if (rdeg) {
        const float r = rdeg[us];
        v.x *= r; v.y *= r;
      }
      float* drow = dst + (long long)ud * D + lane * 2;
      atomicAdd(drow, v.x);      // global_atomic_add_f32, no-return (STOREcnt)
      atomicAdd(drow + 1, v.y);
    }
    __syncthreads();  // all reads of `buf` done before it is re-staged
    buf ^= 1;
  }
}

// -------------------- utility kernels --------------------
__global__ void zero_kernel(float* __restrict__ p, long long n) {
  long long stride = (long long)gridDim.x * blockDim.x * 4;
  for (long long i = ((long long)blockIdx.x * blockDim.x + threadIdx.x) * 4;
       i < n; i += stride) {
    if (i + 3 < n) {
      float4 z; z.x = 0.f; z.y = 0.f; z.z = 0.f; z.w = 0.f;
      *(float4*)(p + i) = z;
    } else {
      for (long long k = i; k < n; ++k) p[k] = 0.f;
    }
  }
}

__global__ void deg_kernel(const int* __restrict__ eu, const int* __restrict__ ei,
                           float* __restrict__ du, float* __restrict__ di, int E) {
  int stride = gridDim.x * blockDim.x;
  for (int e = blockIdx.x * blockDim.x + threadIdx.x; e < E; e += stride) {
    atomicAdd(&du[eu[e]], 1.0f);
    atomicAdd(&di[ei[e]], 1.0f);
  }
}

__global__ void recip_kernel(float* __restrict__ p, int n) {
  int stride = gridDim.x * blockDim.x;
  for (int i = blockIdx.x * blockDim.x + threadIdx.x; i < n; i += stride)
    p[i] = 1.0f / fmaxf(p[i], 1.0f);
}

// x[node][0:64] *= r[node], float4-vectorized (D=64 -> 16 float4 per node)
__global__ void scale_kernel(float* __restrict__ x, const float* __restrict__ r, int nodes) {
  long long total = (long long)nodes * (D / 4);
  long long stride = (long long)gridDim.x * blockDim.x;
  for (long long i = (long long)blockIdx.x * blockDim.x + threadIdx.x; i < total; i += stride) {
    int node = (int)(i / (D / 4));
    float rr = r[node];
    float4* p = (float4*)x + i;
    float4 v = *p;
    v.x *= rr; v.y *= rr; v.z *= rr; v.w *= rr;
    *p = v;
  }
}

// -------------------- host launcher --------------------
extern "C" void kernel_launch(void* const* d_in, const int* in_sizes, int n_in,
                              void* d_out, int out_size, void* d_ws, size_t ws_size,
                              hipStream_t stream) {
  (void)n_in; (void)out_size; (void)ws_size;
  const float* emb_u = (const float*)d_in[0];
  const float* emb_i = (const float*)d_in[1];
  const int*   edge_u = (const int*)d_in[2];
  const int*   edge_i = (const int*)d_in[3];
  const int NU = in_sizes[0] / D;
  const int NI = in_sizes[1] / D;
  const int E  = in_sizes[2];

  float* ws    = (float*)d_ws;
  float* rdeg_u = ws;                      // NU   (holds 1/max(deg,1) after recip)
  float* rdeg_i = rdeg_u + NU;             // NI
  float* huA    = rdeg_i + NI;             // NU*D  layer-1 user sums
  float* hiA    = huA + (long long)NU * D; // NI*D  layer-1 item sums
  float* huB    = hiA + (long long)NI * D; // NU*D  layer-2 user sums
  float* hiB    = huB + (long long)NU * D; // NI*D  layer-2 item sums

  float* out_u = (float*)d_out;
  float* out_i = out_u + (long long)NU * D;

  const int B = 256;
  auto g4 = [](long long n) {
    long long g = (n + 4LL * 256 - 1) / (4LL * 256);
    return (int)(g > 4096 ? 4096 : (g < 1 ? 1 : g));
  };
  auto g1 = [](long long n) {
    long long g = (n + 255) / 256;
    return (int)(g > 4096 ? 4096 : (g < 1 ? 1 : g));
  };

  const long long NUD = (long long)NU * D, NID = (long long)NI * D;
  const int ntiles = (E + TILE - 1) / TILE;
  const int sgrid = ntiles < 2048 ? ntiles : 2048;

  // degrees -> reciprocals
  zero_kernel<<<g4(NU), B, 0, stream>>>(rdeg_u, NU);
  zero_kernel<<<g4(NI), B, 0, stream>>>(rdeg_i, NI);
  deg_kernel<<<2048, B, 0, stream>>>(edge_u, edge_i, rdeg_u, rdeg_i, E);
  recip_kernel<<<g1(NU), B, 0, stream>>>(rdeg_u, NU);
  recip_kernel<<<g1(NI), B, 0, stream>>>(rdeg_i, NI);

  // layer 1: raw embeddings -> unscaled sums
  zero_kernel<<<g4(NUD), B, 0, stream>>>(huA, NUD);
  zero_kernel<<<g4(NID), B, 0, stream>>>(hiA, NID);
  scatter_kernel<<<sgrid, SCATTER_BLOCK, 0, stream>>>(emb_u, nullptr, edge_u, edge_i, hiA, E);
  scatter_kernel<<<sgrid, SCATTER_BLOCK, 0, stream>>>(emb_i, nullptr, edge_i, edge_u, huA, E);

  // layer 2: scale-by-rdeg fused at gather
  zero_kernel<<<g4(NUD), B, 0, stream>>>(huB, NUD);
  zero_kernel<<<g4(NID), B, 0, stream>>>(hiB, NID);
  scatter_kernel<<<sgrid, SCATTER_BLOCK, 0, stream>>>(huA, rdeg_u, edge_u, edge_i, hiB, E);
  scatter_kernel<<<sgrid, SCATTER_BLOCK, 0, stream>>>(hiA, rdeg_i, edge_i, edge_u, huB, E);

  // layer 3 -> d_out (still unscaled sums)
  zero_kernel<<<g4(NUD), B, 0, stream>>>(out_u, NUD);
  zero_kernel<<<g4(NID), B, 0, stream>>>(out_i, NID);
  scatter_kernel<<<sgrid, SCATTER_BLOCK, 0, stream>>>(huB, rdeg_u, edge_u, edge_i, out_i, E);
  scatter_kernel<<<sgrid, SCATTER_BLOCK, 0, stream>>>(hiB, rdeg_i, edge_i, edge_u, out_u, E);

  // final mean scale
  scale_kernel<<<g1(NUD / 4), B, 0, stream>>>(out_u, rdeg_u, NU);
  scale_kernel<<<g1(NID / 4), B, 0, stream>>>(out_i, rdeg_i, NI);
}